// VointGraphConv_19911468384710
// MI455X (gfx1250) — compile-verified
//
#include <hip/hip_runtime.h>
#include <hip/hip_bf16.h>

// VointGraphConv forward, algebraically collapsed:
//   y[b,o,n,v,w] = u[b,o,n,v] + z[b,o,n,(v-w)%V],  u=(W1-W2)@xn, z=W2@xn
// so the VxV expansion, its BN stats, and the masked max all have closed forms.
// Channel GEMMs run on the CDNA5 tensor pipe via V_WMMA_F32_16X16X4_F32 (fp32).

typedef __attribute__((ext_vector_type(2))) float v2f;
typedef __attribute__((ext_vector_type(8))) float v8f;

#define BB 4
#define CC 64
#define C2 32
#define NN 1024
#define VV 20
#define NVP (NN * VV)        /* 20480 elements per (b, channel) plane   */
#define COLS (BB * NVP)      /* 81920 total (b,n,v) columns             */
#define PTS (BB * NN)        /* 4096 points                             */

// ---------------------------------------------------------------- utilities
__global__ void vgc_zero(float* p, int n) {
  int i = blockIdx.x * blockDim.x + threadIdx.x;
  if (i < n) p[i] = 0.f;
}

// ------------------------------------------------- kernel 1: embed GEMM (WMMA)
// x_s[b,o,n,v] = (sum_c w_embed[o,c]*feats[b,c,n,v] + b_embed[o]) * mask
// one wave per 16-column tile; M = 2 tiles of 16 (32 out ch); K = 64 = 16x4.
__global__ void vgc_embed(const float* __restrict__ feats,
                          const float* __restrict__ mask,
                          const float* __restrict__ w_embed,
                          const float* __restrict__ b_embed,
                          float* __restrict__ xs) {
  int wave = (blockIdx.x * blockDim.x + threadIdx.x) >> 5;
  int lane = threadIdx.x & 31;
  int colBase = wave << 4;                 // global column of tile start
  int b = colBase / NVP;                   // tiles never straddle b (NVP%16==0)
  int r = colBase - b * NVP;               // column offset inside batch plane
  int half = lane >> 4;                    // selects K-pair (ISA A/B layout)
  int cl = lane & 15;                      // M for A-frag, N for B/C-frag
  int col_r = r + cl;

  const float* fb = feats + (size_t)b * CC * NVP + col_r;
  v8f acc[2] = {};
  for (int kc = 0; kc < 16; ++kc) {
    int c0 = kc * 4 + half * 2;
    v2f bf;
    bf.x = fb[(size_t)c0 * NVP];
    bf.y = fb[(size_t)(c0 + 1) * NVP];
    for (int mt = 0; mt < 2; ++mt) {
      int o = mt * 16 + cl;
      v2f af;
      af.x = w_embed[o * CC + c0];
      af.y = w_embed[o * CC + c0 + 1];
      acc[mt] = __builtin_amdgcn_wmma_f32_16x16x4_f32(
          false, af, false, bf, (short)0, acc[mt], false, false);
    }
  }
  float mk = mask[colBase + cl];
  for (int mt = 0; mt < 2; ++mt)
    for (int rr = 0; rr < 8; ++rr) {
      int o = mt * 16 + rr + half * 8;
      float val = (acc[mt][rr] + b_embed[o]) * mk;
      xs[(size_t)(b * C2 + o) * NVP + col_r] = val;
    }
}

// ------------------------------------------- kernel 2: per-channel x statistics
// one block per (b, ch) plane; st[0..31]=sum, st[32..63]=sumsq
__global__ void vgc_xstats(const float* __restrict__ xs, float* __restrict__ st) {
  int plane = blockIdx.x;                  // b*C2 + ch
  int ch = plane % C2;
  const float* p = xs + (size_t)plane * NVP;
  float s = 0.f, s2 = 0.f;
  for (int i = threadIdx.x; i < NVP; i += blockDim.x) {
    float v = p[i];
    s += v; s2 += v * v;
  }
  __shared__ float red[2];
  if (threadIdx.x == 0) { red[0] = 0.f; red[1] = 0.f; }
  __syncthreads();
  atomicAdd(&red[0], s);
  atomicAdd(&red[1], s2);
  __syncthreads();
  if (threadIdx.x == 0) {
    atomicAdd(&st[ch], red[0]);
    atomicAdd(&st[C2 + ch], red[1]);
  }
}

// ----------------------------------------------- kernel 3: embed BN param fold
__global__ void vgc_embed_bn(const float* __restrict__ st,
                             const float* __restrict__ g,
                             const float* __restrict__ be,
                             float* __restrict__ prm) {
  int c = threadIdx.x;
  if (c >= C2) return;
  float inv = 1.f / (float)COLS;
  float m = st[c] * inv;
  float var = st[C2 + c] * inv - m * m;
  float sc = g[c] * rsqrtf(var + 1e-5f);
  prm[c] = sc;             // scale
  prm[C2 + c] = be[c] - m * sc;  // shift
}

// --------------------------------------- kernel 4: conv GEMMs u,z (WMMA, fused)
// xn = relu(scale*x_s + shift) * mask applied on load; one wave per 16 columns.
__global__ void vgc_conv(const float* __restrict__ xs,
                         const float* __restrict__ mask,
                         const float* __restrict__ w_conv,
                         const float* __restrict__ prm,
                         float* __restrict__ u, float* __restrict__ z) {
  int wave = (blockIdx.x * blockDim.x + threadIdx.x) >> 5;
  int lane = threadIdx.x & 31;
  int colBase = wave << 4;
  int b = colBase / NVP;
  int r = colBase - b * NVP;
  int half = lane >> 4;
  int cl = lane & 15;
  int col_r = r + cl;
  float mk = mask[colBase + cl];

  const float* xb = xs + (size_t)b * C2 * NVP + col_r;
  v2f bfr[8];
  for (int kc = 0; kc < 8; ++kc) {
    int c0 = kc * 4 + half * 2;
    float x0 = xb[(size_t)c0 * NVP];
    float x1 = xb[(size_t)(c0 + 1) * NVP];
    x0 = fmaxf(prm[c0] * x0 + prm[C2 + c0], 0.f) * mk;
    x1 = fmaxf(prm[c0 + 1] * x1 + prm[C2 + c0 + 1], 0.f) * mk;
    bfr[kc].x = x0;
    bfr[kc].y = x1;
  }

  v8f accU[4] = {};
  v8f accZ[4] = {};
  for (int kc = 0; kc < 8; ++kc) {
    int c0 = kc * 4 + half * 2;
    for (int mt = 0; mt < 4; ++mt) {
      int o = mt * 16 + cl;
      float w1a = w_conv[o * CC + c0];
      float w1b = w_conv[o * CC + c0 + 1];
      float w2a = w_conv[o * CC + 32 + c0];
      float w2b = w_conv[o * CC + 32 + c0 + 1];
      v2f aU; aU.x = w1a - w2a; aU.y = w1b - w2b;
      v2f aZ; aZ.x = w2a;       aZ.y = w2b;
      accU[mt] = __builtin_amdgcn_wmma_f32_16x16x4_f32(
          false, aU, false, bfr[kc], (short)0, accU[mt], false, false);
      accZ[mt] = __builtin_amdgcn_wmma_f32_16x16x4_f32(
          false, aZ, false, bfr[kc], (short)0, accZ[mt], false, false);
    }
  }
  for (int mt = 0; mt < 4; ++mt)
    for (int rr = 0; rr < 8; ++rr) {
      int o = mt * 16 + rr + half * 8;
      size_t idx = (size_t)(b * CC + o) * NVP + col_r;
      u[idx] = accU[mt][rr];
      z[idx] = accZ[mt][rr];
    }
}

// --------------------- kernel 5: closed-form BN stats for y from u,z aggregates
// st: [0,64)=S1u [64,128)=S1z [128,192)=S2u [192,256)=S2z [256,320)=Sc
__global__ void vgc_ystats(const float* __restrict__ u,
                           const float* __restrict__ z,
                           float* __restrict__ st) {
  __shared__ float ls[320];
  for (int i = threadIdx.x; i < 320; i += blockDim.x) ls[i] = 0.f;
  __syncthreads();
  int o = threadIdx.x & 63;
  int ps = threadIdx.x >> 6;               // 0..3
  float a1 = 0.f, a2 = 0.f, a3 = 0.f, a4 = 0.f, a5 = 0.f;
  for (int j = 0; j < 4; ++j) {
    int p = blockIdx.x * 16 + ps * 4 + j;  // point (b,n)
    int b = p >> 10;
    int n = p & (NN - 1);
    size_t base = (size_t)(b * CC + o) * NVP + n * VV;
    float su = 0.f, sz = 0.f;
    for (int v = 0; v < VV; ++v) {
      float uu = u[base + v];
      float zz = z[base + v];
      su += uu; sz += zz;
      a3 += uu * uu; a4 += zz * zz;
    }
    a1 += su; a2 += sz; a5 += su * sz;
  }
  atomicAdd(&ls[o], a1);
  atomicAdd(&ls[64 + o], a2);
  atomicAdd(&ls[128 + o], a3);
  atomicAdd(&ls[192 + o], a4);
  atomicAdd(&ls[256 + o], a5);
  __syncthreads();
  for (int i = threadIdx.x; i < 320; i += blockDim.x) atomicAdd(&st[i], ls[i]);
}

// ------------------------------------------------ kernel 6: conv BN param fold
__global__ void vgc_conv_bn(const float* __restrict__ st,
                            const float* __restrict__ g,
                            const float* __restrict__ be,
                            float* __restrict__ prm) {
  int o = threadIdx.x;
  if (o >= CC) return;
  float inv1 = 1.f / (float)COLS;          // 1/(B*N*V)
  float mean = (st[o] + st[64 + o]) * inv1;
  float e2 = (st[128 + o] + st[192 + o]) * inv1
           + 2.f * st[256 + o] * inv1 / (float)VV;
  float var = e2 - mean * mean;
  float sc = g[o] * rsqrtf(var + 1e-5f);
  prm[o] = sc;
  prm[CC + o] = be[o] - mean * sc;
}

// ---------------- kernel 7: fused max-over-w + relu + mask + residual (output)
__global__ void vgc_final(const float* __restrict__ feats,
                          const float* __restrict__ mask,
                          const float* __restrict__ u,
                          const float* __restrict__ z,
                          const float* __restrict__ prm,
                          float* __restrict__ out) {
  int o = threadIdx.x & 63;
  int pl = threadIdx.x >> 6;
  int p = blockIdx.x * 4 + pl;             // point (b,n)
  int b = p >> 10;
  int n = p & (NN - 1);
  size_t base = (size_t)(b * CC + o) * NVP + n * VV;
  size_t mbase = (size_t)p * VV;
  float sc = prm[o], tc = prm[CC + o];
  float zmax = -1e38f, zmin = 1e38f;
  for (int v = 0; v < VV; ++v) {
    float zz = z[base + v];
    zmax = fmaxf(zmax, zz);
    zmin = fminf(zmin, zz);
  }
  float zterm = (sc >= 0.f) ? sc * zmax : sc * zmin;
  for (int v = 0; v < VV; ++v) {
    float mk = mask[mbase + v];
    float val = fmaxf(sc * u[base + v] + tc + zterm, 0.f) * mk;
    out[base + v] = feats[base + v] + val;
  }
}

// -------------------------------------------------------------------- launcher
extern "C" void kernel_launch(void* const* d_in, const int* in_sizes, int n_in,
                              void* d_out, int out_size, void* d_ws, size_t ws_size,
                              hipStream_t stream) {
  const float* feats   = (const float*)d_in[0];
  const float* mask    = (const float*)d_in[1];
  const float* w_embed = (const float*)d_in[2];
  const float* b_embed = (const float*)d_in[3];
  const float* g_embed = (const float*)d_in[4];
  const float* be_embed= (const float*)d_in[5];
  const float* w_conv  = (const float*)d_in[6];
  const float* g_conv  = (const float*)d_in[7];
  const float* be_conv = (const float*)d_in[8];
  float* out = (float*)d_out;

  float* ws = (float*)d_ws;
  const size_t XN = (size_t)BB * C2 * NVP;   // x_s
  const size_t UN = (size_t)BB * CC * NVP;   // u and z each
  float* xs   = ws;
  float* u    = ws + XN;
  float* z    = ws + XN + UN;
  float* stE  = ws + XN + 2 * UN;            // 64 floats
  float* prmE = stE + 64;                    // 64 floats
  float* stY  = stE + 128;                   // 320 floats
  float* prmY = stE + 448;                   // 128 floats

  // zero stat accumulators (576 floats covers stE..prmY)
  vgc_zero<<<3, 256, 0, stream>>>(stE, 576);

  // embed GEMM: 5120 column tiles, 8 waves/block
  vgc_embed<<<640, 256, 0, stream>>>(feats, mask, w_embed, b_embed, xs);
  vgc_xstats<<<BB * C2, 256, 0, stream>>>(xs, stE);
  vgc_embed_bn<<<1, 32, 0, stream>>>(stE, g_embed, be_embed, prmE);

  // conv GEMMs u,z
  vgc_conv<<<640, 256, 0, stream>>>(xs, mask, w_conv, prmY /*unused*/ == prmY ? prmE : prmE, u, z);
  vgc_ystats<<<PTS / 16, 256, 0, stream>>>(u, z, stY);
  vgc_conv_bn<<<1, 64, 0, stream>>>(stY, g_conv, be_conv, prmY);

  // fused epilogue
  vgc_final<<<PTS / 4, 256, 0, stream>>>(feats, mask, u, z, prmY, out);
}